// EncoderLayer_29059748725632
// MI455X (gfx1250) — compile-verified
//
#include <hip/hip_runtime.h>

#define N_NODES 20000
#define N_EDGES 150000
#define D 512
#define BN_EPS 1e-5f

typedef __attribute__((ext_vector_type(2))) float v2f;
typedef __attribute__((ext_vector_type(8))) float v8f;

// ---------------------------------------------------------------- zero scratch
__global__ void zero_kernel(float* __restrict__ p, long long n) {
  long long i = (long long)blockIdx.x * blockDim.x + threadIdx.x;
  long long stride = (long long)gridDim.x * blockDim.x;
  for (; i < n; i += stride) p[i] = 0.0f;
}

// ------------------------------------------------- edge scatter: agg += x[src]
// one wave32 per edge; each lane handles 4 float4 (16 floats) of the 512-dim row
__global__ __launch_bounds__(256)
void scatter_kernel(const float* __restrict__ x,
                    const int* __restrict__ src,
                    const int* __restrict__ dst,
                    float* __restrict__ agg,
                    float* __restrict__ indeg) {
  int wave = threadIdx.x >> 5;
  int lane = threadIdx.x & 31;
  int e = blockIdx.x * 8 + wave;
  if (e >= N_EDGES) return;
  int s = src[e];
  int d = dst[e];
  const float4* xs = (const float4*)(x + (size_t)s * D);
  float* ag = agg + (size_t)d * D;
  if (lane == 0) atomicAdd(indeg + d, 1.0f);
#pragma unroll
  for (int j = 0; j < 4; ++j) {
    int idx = j * 32 + lane;            // float4 index 0..127, lane-consecutive
    float4 v = xs[idx];
    atomicAdd(ag + idx * 4 + 0, v.x);
    atomicAdd(ag + idx * 4 + 1, v.y);
    atomicAdd(ag + idx * 4 + 2, v.z);
    atomicAdd(ag + idx * 4 + 3, v.w);
  }
}

// ------------------------- fused select + GEMM(WMMA f32) + bias + ReLU + stats
// block = 256 threads (8 wave32), computes a 128x64 output tile.
// A = h (rows picked from agg or x per indeg), B = W^T (B[k][n] = W[n][k]).
__global__ __launch_bounds__(256)
void gemm_wmma_kernel(const float* __restrict__ x,
                      const float* __restrict__ agg,
                      const float* __restrict__ indeg,
                      const float* __restrict__ Wm,
                      const float* __restrict__ bias,
                      float* __restrict__ out,
                      float* __restrict__ colsum,
                      float* __restrict__ colsumsq) {
  __shared__ float Alds[128 * 33];   // 128 rows x 32 K, +1 pad word
  __shared__ float Blds[64 * 33];    // 64 cols x 32 K, +1 pad word

  const int t = threadIdx.x;
  const int lane = t & 31;
  const int wave = t >> 5;
  const int rowTile = blockIdx.y * 128;
  const int colTile = blockIdx.x * 64;
  const int kg = (t & 7) * 4;        // this thread's K sub-offset within a 32-chunk

  // Per-thread staging pointers: 4 A-rows, 2 B-cols (fixed across K steps)
  const float* asrc[4];
#pragma unroll
  for (int i = 0; i < 4; ++i) {
    int r = (t >> 3) + 32 * i;       // local row 0..127
    int g = rowTile + r;
    asrc[i] = (g < N_NODES) ? ((indeg[g] > 0.0f ? agg : x) + (size_t)g * D) : nullptr;
  }
  const float* bsrc[2];
#pragma unroll
  for (int i = 0; i < 2; ++i) {
    int c = (t >> 3) + 32 * i;       // local col 0..63
    bsrc[i] = Wm + (size_t)(colTile + c) * D + kg;
  }

  v8f acc0 = {}, acc1 = {}, acc2 = {}, acc3 = {};

  for (int kt = 0; kt < D; kt += 32) {
    __syncthreads();
    // stage A (coalesced float4 along K)
#pragma unroll
    for (int i = 0; i < 4; ++i) {
      int r = (t >> 3) + 32 * i;
      float4 v = make_float4(0.f, 0.f, 0.f, 0.f);
      if (asrc[i]) v = *(const float4*)(asrc[i] + kt + kg);
      Alds[r * 33 + kg + 0] = v.x;
      Alds[r * 33 + kg + 1] = v.y;
      Alds[r * 33 + kg + 2] = v.z;
      Alds[r * 33 + kg + 3] = v.w;
    }
    // stage B (W rows are contiguous in K -> coalesced)
#pragma unroll
    for (int i = 0; i < 2; ++i) {
      int c = (t >> 3) + 32 * i;
      float4 v = *(const float4*)(bsrc[i] + kt);
      Blds[c * 33 + kg + 0] = v.x;
      Blds[c * 33 + kg + 1] = v.y;
      Blds[c * 33 + kg + 2] = v.z;
      Blds[c * 33 + kg + 3] = v.w;
    }
    __syncthreads();

    // ISA 16x4 f32 A layout: lanes 0-15 = M, VGPR0/1 = K,K+1; lanes 16-31 = K+2,K+3
    const int arow = wave * 16 + (lane & 15);
    const int bcol = lane & 15;
    const int khalf = (lane >> 4) * 2;
#pragma unroll
    for (int kk = 0; kk < 32; kk += 4) {
      int ko = kk + khalf;
      v2f a;  a[0]  = Alds[arow * 33 + ko];        a[1]  = Alds[arow * 33 + ko + 1];
      v2f b0; b0[0] = Blds[(bcol +  0) * 33 + ko]; b0[1] = Blds[(bcol +  0) * 33 + ko + 1];
      v2f b1; b1[0] = Blds[(bcol + 16) * 33 + ko]; b1[1] = Blds[(bcol + 16) * 33 + ko + 1];
      v2f b2; b2[0] = Blds[(bcol + 32) * 33 + ko]; b2[1] = Blds[(bcol + 32) * 33 + ko + 1];
      v2f b3; b3[0] = Blds[(bcol + 48) * 33 + ko]; b3[1] = Blds[(bcol + 48) * 33 + ko + 1];
      acc0 = __builtin_amdgcn_wmma_f32_16x16x4_f32(false, a, false, b0, (short)0, acc0, false, false);
      acc1 = __builtin_amdgcn_wmma_f32_16x16x4_f32(false, a, false, b1, (short)0, acc1, false, false);
      acc2 = __builtin_amdgcn_wmma_f32_16x16x4_f32(false, a, false, b2, (short)0, acc2, false, false);
      acc3 = __builtin_amdgcn_wmma_f32_16x16x4_f32(false, a, false, b3, (short)0, acc3, false, false);
    }
  }

  // Epilogue: bias + ReLU + store + column sum/sumsq for BN.
  // C/D layout: VGPR r -> M = r (lanes 0-15) / r+8 (lanes 16-31), N = lane&15.
  int gRowBase = rowTile + wave * 16 + ((lane >> 4) << 3);
  v8f accs[4] = {acc0, acc1, acc2, acc3};
#pragma unroll
  for (int nt = 0; nt < 4; ++nt) {
    int col = colTile + nt * 16 + (lane & 15);
    float bv = bias[col];
    float ps = 0.0f, pq = 0.0f;
#pragma unroll
    for (int r = 0; r < 8; ++r) {
      int grow = gRowBase + r;
      float h = accs[nt][r] + bv;
      h = h > 0.0f ? h : 0.0f;
      if (grow < N_NODES) {
        out[(size_t)grow * D + col] = h;
        ps += h;
        pq += h * h;
      }
    }
    // fold the two 16-lane halves (same col, different M ranges)
    ps += __shfl_xor(ps, 16, 32);
    pq += __shfl_xor(pq, 16, 32);
    if (lane < 16) {
      atomicAdd(colsum + col, ps);
      atomicAdd(colsumsq + col, pq);
    }
  }
}

// ------------------------------------------------------------ BN normalization
__global__ void bn_kernel(float* __restrict__ out,
                          const float* __restrict__ colsum,
                          const float* __restrict__ colsumsq,
                          const float* __restrict__ gamma,
                          const float* __restrict__ beta) {
  const float invN = 1.0f / (float)N_NODES;
  long long total = (long long)N_NODES * D;
  long long i = (long long)blockIdx.x * blockDim.x + threadIdx.x;
  long long stride = (long long)gridDim.x * blockDim.x;
  for (; i < total; i += stride) {
    int col = (int)(i & (D - 1));
    float mean = colsum[col] * invN;
    float var = colsumsq[col] * invN - mean * mean;
    out[i] = (out[i] - mean) * rsqrtf(var + BN_EPS) * gamma[col] + beta[col];
  }
}

// ------------------------------------------------------------------------ host
extern "C" void kernel_launch(void* const* d_in, const int* in_sizes, int n_in,
                              void* d_out, int out_size, void* d_ws, size_t ws_size,
                              hipStream_t stream) {
  const float* x     = (const float*)d_in[0];
  const int*   src   = (const int*)d_in[1];
  const int*   dst   = (const int*)d_in[2];
  const float* Wm    = (const float*)d_in[3];
  const float* bias  = (const float*)d_in[4];
  const float* gamma = (const float*)d_in[5];
  const float* beta  = (const float*)d_in[6];
  float* out = (float*)d_out;

  float* ws       = (float*)d_ws;
  float* agg      = ws;                               // N_NODES * D
  float* indeg    = agg + (size_t)N_NODES * D;        // N_NODES
  float* colsum   = indeg + N_NODES;                  // D
  float* colsumsq = colsum + D;                       // D

  long long zcount = (long long)N_NODES * D + N_NODES + 2 * D;
  zero_kernel<<<2048, 256, 0, stream>>>(ws, zcount);

  scatter_kernel<<<(N_EDGES + 7) / 8, 256, 0, stream>>>(x, src, dst, agg, indeg);

  dim3 grid(D / 64, (N_NODES + 127) / 128);
  gemm_wmma_kernel<<<grid, 256, 0, stream>>>(x, agg, indeg, Wm, bias, out,
                                             colsum, colsumsq);

  bn_kernel<<<4096, 256, 0, stream>>>(out, colsum, colsumsq, gamma, beta);
}